// WeightedGraphGNN_180388626680
// MI455X (gfx1250) — compile-verified
//
#include <hip/hip_runtime.h>

#define N_NODES 50000
#define N_EDGES 600000
#define D 128
#define NODE_STRIDE 50048  // 64-float aligned padding for per-node arrays

typedef __attribute__((ext_vector_type(2))) float v2f;
typedef __attribute__((ext_vector_type(8))) float v8f;

// ---------------------------------------------------------------- utilities
__global__ void fill_zero(float* __restrict__ p, int n) {
    int i = blockIdx.x * blockDim.x + threadIdx.x;
    if (i < n) p[i] = 0.0f;
}

__global__ void deg_accum(const int* __restrict__ esrc, const int* __restrict__ edst,
                          float* __restrict__ deg_src, float* __restrict__ deg_dst, int E) {
    int e = blockIdx.x * blockDim.x + threadIdx.x;
    if (e < E) {
        atomicAdd(&deg_src[esrc[e]], 1.0f);
        atomicAdd(&deg_dst[edst[e]], 1.0f);
    }
}

// deg -> 1/sqrt(max(deg,1)), in place (covers both arrays + padding; padding is 0 -> 1.0)
__global__ void deg_to_norm(float* __restrict__ p, int n) {
    int i = blockIdx.x * blockDim.x + threadIdx.x;
    if (i < n) {
        float d = p[i];
        d = d < 1.0f ? 1.0f : d;
        p[i] = 1.0f / sqrtf(d);
    }
}

// ------------------------------------------------- edge gather + scatter-add
// One wave (32 lanes) per edge; each lane handles 4 of the 128 feature cols.
// Folds norm_src[src] * edge_weight into the gathered value.
__global__ void scatter_edges(const float* __restrict__ hin,
                              const float* __restrict__ norm_src,
                              const float* __restrict__ ew,
                              const int* __restrict__ esrc,
                              const int* __restrict__ edst,
                              float* __restrict__ agg, int E) {
    int e    = blockIdx.x * (blockDim.x >> 5) + (threadIdx.x >> 5);
    int lane = threadIdx.x & 31;
    if (e >= E) return;
    int s = esrc[e];
    int d = edst[e];
    float w = ew[e] * norm_src[s];
    const float4* in = (const float4*)(hin + (size_t)s * D) + lane;
    float4 v = *in;
    float* out = agg + (size_t)d * D + lane * 4;
    atomicAdd(out + 0, v.x * w);
    atomicAdd(out + 1, v.y * w);
    atomicAdd(out + 2, v.z * w);
    atomicAdd(out + 3, v.w * w);
}

// ----------------------------------------------------------- WMMA f32 GEMM
// out[M,N] = relu?( (scale[M]*A[M,:]) @ W + bias )  for a 16x16 tile per wave.
// A: [n_rows x 128] (agg, scaled per-row by norm_dst), W: [128 x 128] row-major.
// Uses V_WMMA_F32_16X16X4_F32, 32 K-steps of 4.
//
// Fragment layouts (ISA 7.12.2, 32-bit, wave32):
//   A 16x4:  lane<16: M=lane,   v0=K0, v1=K1 ; lane>=16: M=lane-16, v0=K2, v1=K3
//   B 4x16:  lane<16: N=lane,   v0=K0, v1=K1 ; lane>=16: N=lane-16, v0=K2, v1=K3
//   C 16x16: VGPR r:  lane<16 -> row r, col lane ; lane>=16 -> row r+8, col lane-16
__global__ void gemm_tile(const float* __restrict__ A,
                          const float* __restrict__ scale,
                          const float* __restrict__ W,
                          const float* __restrict__ bias,
                          float* __restrict__ out, int n_rows, int relu) {
    int waves_per_block = blockDim.x >> 5;
    int rb   = blockIdx.x * waves_per_block + (threadIdx.x >> 5);
    int n_rb = n_rows >> 4;
    if (rb >= n_rb) return;

    int lane = threadIdx.x & 31;
    int half = lane >> 4;   // 0: K-pair {0,1}, 1: K-pair {2,3} within each K4 step
    int n    = lane & 15;
    int col0 = blockIdx.y << 4;

    // Preload all B fragments for this 16-wide column tile (kept in registers).
    v2f bfrag[32];
#pragma unroll
    for (int k = 0; k < 32; ++k) {
        int kk = k * 4 + half * 2;
        bfrag[k].x = W[(kk + 0) * D + col0 + n];
        bfrag[k].y = W[(kk + 1) * D + col0 + n];
    }

    int row0 = rb << 4;
    int row  = row0 + n;                      // A row for this lane (both halves)
    float s  = scale[row];
    const float* arow = A + (size_t)row * D + half * 2;

    v8f c = {};
#pragma unroll
    for (int k = 0; k < 32; ++k) {
        v2f a = *(const v2f*)(arow + k * 4);  // 8B aligned (even offsets)
        a.x *= s;
        a.y *= s;
        c = __builtin_amdgcn_wmma_f32_16x16x4_f32(
            /*neg_a=*/false, a, /*neg_b=*/false, bfrag[k],
            /*c_mod=*/(short)0, c, /*reuse_a=*/false, /*reuse_b=*/false);
    }

    float bv = bias[col0 + n];
#pragma unroll
    for (int r = 0; r < 8; ++r) {
        float v = c[r] + bv;
        if (relu) v = v > 0.0f ? v : 0.0f;
        out[(size_t)(row0 + half * 8 + r) * D + col0 + n] = v;
    }
}

// ------------------------------------------------------------------ launch
extern "C" void kernel_launch(void* const* d_in, const int* in_sizes, int n_in,
                              void* d_out, int out_size, void* d_ws, size_t ws_size,
                              hipStream_t stream) {
    (void)in_sizes; (void)n_in; (void)out_size; (void)ws_size;

    const float* features = (const float*)d_in[0];
    const float* ew       = (const float*)d_in[1];
    const float* W1       = (const float*)d_in[2];
    const float* b1       = (const float*)d_in[3];
    const float* W2       = (const float*)d_in[4];
    const float* b2       = (const float*)d_in[5];
    const int*   esrc     = (const int*)d_in[6];
    const int*   edst     = (const int*)d_in[7];
    float*       out      = (float*)d_out;

    // Workspace carve: [deg/norm_src | deg/norm_dst | agg | h1]  (~51.6 MB)
    float* deg_src = (float*)d_ws;
    float* deg_dst = deg_src + NODE_STRIDE;
    float* agg     = deg_dst + NODE_STRIDE;
    float* h1      = agg + (size_t)N_NODES * D;

    // 1) zero degree arrays + agg (ws is poisoned; must re-init every call)
    int zero_n = 2 * NODE_STRIDE + N_NODES * D;
    fill_zero<<<(zero_n + 255) / 256, 256, 0, stream>>>(deg_src, zero_n);

    // 2) degrees -> norms
    deg_accum<<<(N_EDGES + 255) / 256, 256, 0, stream>>>(esrc, edst, deg_src, deg_dst, N_EDGES);
    deg_to_norm<<<(2 * NODE_STRIDE + 255) / 256, 256, 0, stream>>>(deg_src, 2 * NODE_STRIDE);

    // 3) layer 1: gather/scatter then WMMA GEMM (+bias, ReLU)
    scatter_edges<<<(N_EDGES + 7) / 8, 256, 0, stream>>>(features, deg_src, ew, esrc, edst, agg, N_EDGES);
    dim3 ggrid((3125 + 3) / 4, 8);  // 50000/16 row blocks x 8 col tiles, 4 waves/block
    gemm_tile<<<ggrid, 128, 0, stream>>>(agg, deg_dst, W1, b1, h1, N_NODES, 1);

    // 4) layer 2
    fill_zero<<<((N_NODES * D) + 255) / 256, 256, 0, stream>>>(agg, N_NODES * D);
    scatter_edges<<<(N_EDGES + 7) / 8, 256, 0, stream>>>(h1, deg_src, ew, esrc, edst, agg, N_EDGES);
    gemm_tile<<<ggrid, 128, 0, stream>>>(agg, deg_dst, W2, b2, out, N_NODES, 0);
}